// QuantumFraudDetector_65481071404825
// MI455X (gfx1250) — compile-verified
//
#include <hip/hip_runtime.h>

typedef __attribute__((ext_vector_type(2))) float v2f;
typedef __attribute__((ext_vector_type(8))) float v8f;

#define NQ      12
#define DIM     4096      // 2^12
#define NS      4         // samples resident in LDS per workgroup (128KB)
#define NL      3         // layers
#define THREADS 256       // 8 wave32 waves

// Accumulated CNOT-chain permutation after L layers (GF(2)-linear gather map):
// g(z)=z^(z>>1); g^2=z^(z>>2); g^3=z^(z>>1)^(z>>2)^(z>>3). L is compile-time.
template <int L>
__device__ __forceinline__ int permc(int z) {
  return (L == 0) ? z
       : (L == 1) ? (z ^ (z >> 1))
       : (L == 2) ? (z ^ (z >> 2))
                  : (z ^ (z >> 1) ^ (z >> 2) ^ (z >> 3));
}

// ---------------------------------------------------------------------------
// Precompute, for each (layer, wire-group-of-4), the fused 16x16 complex gate:
//   U = g_w ⊗ g_{w+1} ⊗ g_{w+2} ⊗ g_{w+3},  g = RY(p2)·RZ(p1)·RY(p0)
// ---------------------------------------------------------------------------
__global__ void qgates_kernel(const float* __restrict__ params,
                              float* __restrict__ Ure, float* __restrict__ Uim) {
  const int t = threadIdx.x;           // 256 threads: one (r,c) entry each
  const int r = t >> 4, c = t & 15;
  for (int l = 0; l < NL; ++l) {
    for (int g = 0; g < 3; ++g) {
      float gr[4][2][2], gi[4][2][2];
      #pragma unroll
      for (int j = 0; j < 4; ++j) {
        const int wire = g * 4 + j;
        const float* p = params + (l * NQ + wire) * 3;
        const float c0 = cosf(0.5f * p[0]), s0 = sinf(0.5f * p[0]);
        const float cp = cosf(0.5f * p[1]), sp = sinf(0.5f * p[1]);
        const float c2 = cosf(0.5f * p[2]), s2 = sinf(0.5f * p[2]);
        // g = RY(p2) * diag(e^{-i p1/2}, e^{+i p1/2}) * RY(p0)
        gr[j][0][0] =  cp * (c2 * c0 - s2 * s0);
        gi[j][0][0] = -sp * (c2 * c0 + s2 * s0);
        gr[j][0][1] = -cp * (c2 * s0 + s2 * c0);
        gi[j][0][1] =  sp * (c2 * s0 - s2 * c0);
        gr[j][1][0] =  cp * (s2 * c0 + c2 * s0);
        gi[j][1][0] =  sp * (c2 * s0 - s2 * c0);
        gr[j][1][1] =  cp * (c2 * c0 - s2 * s0);
        gi[j][1][1] =  sp * (c2 * c0 + s2 * s0);
      }
      float ar = 1.f, ai = 0.f;            // Kron-product entry (r,c)
      #pragma unroll
      for (int j = 0; j < 4; ++j) {
        const int rb = (r >> (3 - j)) & 1, cb = (c >> (3 - j)) & 1;
        const float br = gr[j][rb][cb], bi = gi[j][rb][cb];
        const float nr = ar * br - ai * bi, ni = ar * bi + ai * br;
        ar = nr; ai = ni;
      }
      Ure[(l * 3 + g) * 256 + r * 16 + c] = ar;
      Uim[(l * 3 + g) * 256 + r * 16 + c] = ai;
    }
  }
}

// ---------------------------------------------------------------------------
// Apply one 16x16 complex group gate (layer L, wire group G) to all NS states
// in LDS. Fully compile-time L/G: permutation constants fold into literals,
// each LDS address is one v_xor. 16 chained V_WMMA_F32_16X16X4_F32 per tile.
// ---------------------------------------------------------------------------
template <int L, int G>
__device__ __forceinline__ void apply_group(float2* __restrict__ sAmp,
                                            const float* __restrict__ Ure,
                                            const float* __restrict__ Uim,
                                            int lane, int wave) {
  constexpr int RS = (G == 0) ? 256 : (G == 1) ? 16 : 1;  // nibble bit stride
  const float* ur = Ure + (L * 3 + G) * 256;
  const float* ui = Uim + (L * 3 + G) * 256;
  const bool hi = (lane >= 16);          // A/B frags: lanes 16-31 hold K+2,K+3
  const int koff = hi ? 2 : 0;
  const int moff = lane & 15;            // A row M
  const int nn   = lane & 15;            // B/D column within tile

  // A fragments (gate, row-major [M][K])
  v2f aRe[4], aIm[4], aMim[4];
  #pragma unroll
  for (int f = 0; f < 4; ++f) {
    const int k0 = 4 * f + koff;
    v2f tr, ti;
    tr.x = ur[moff * 16 + k0]; tr.y = ur[moff * 16 + k0 + 1];
    ti.x = ui[moff * 16 + k0]; ti.y = ui[moff * 16 + k0 + 1];
    aRe[f] = tr; aIm[f] = ti; aMim[f] = -ti;
  }

  // Per-lane permuted row offsets: perm(k*RS) are compile-time literals.
  int pk0[4], pk1[4], pm[8];
  #pragma unroll
  for (int f = 0; f < 4; ++f) {
    pk0[f] = hi ? permc<L>((4 * f + 2) * RS) : permc<L>((4 * f + 0) * RS);
    pk1[f] = hi ? permc<L>((4 * f + 3) * RS) : permc<L>((4 * f + 1) * RS);
  }
  #pragma unroll
  for (int v = 0; v < 8; ++v)            // D rows: v / v+8 per lane half
    pm[v] = hi ? permc<L>((v + 8) * RS) : permc<L>(v * RS);

  // NS*16 = 64 column-tiles per group, 8 per wave.
  #pragma unroll 2
  for (int t = 0; t < 8; ++t) {
    const int tt = wave * 8 + t;
    const int ss = tt >> 4;
    const int c  = ((tt & 15) << 4) + nn;          // global column 0..255
    int other;                                     // non-nibble bits of z
    if (G == 0)      other = c;
    else if (G == 1) other = ((c >> 4) << 8) | (c & 15);
    else             other = c << 4;
    // perm is GF(2)-linear over disjoint bit fields: combine once per tile.
    const int pob = (ss << 12) | permc<L>(other);

    v2f bRe[4], bIm[4];
    #pragma unroll
    for (int f = 0; f < 4; ++f) {
      const float2 q0 = sAmp[pob ^ pk0[f]];        // ds_load_b64
      const float2 q1 = sAmp[pob ^ pk1[f]];
      v2f tr, ti;
      tr.x = q0.x; tr.y = q1.x;
      ti.x = q0.y; ti.y = q1.y;
      bRe[f] = tr; bIm[f] = ti;
    }
    v8f dre = {}; v8f dim_ = {};
    #pragma unroll
    for (int f = 0; f < 4; ++f) {     // complex matmul: 4 real WMMAs / k-step
      dre  = __builtin_amdgcn_wmma_f32_16x16x4_f32(false, aRe[f],  false, bRe[f], (short)0, dre,  false, false);
      dre  = __builtin_amdgcn_wmma_f32_16x16x4_f32(false, aMim[f], false, bIm[f], (short)0, dre,  false, false);
      dim_ = __builtin_amdgcn_wmma_f32_16x16x4_f32(false, aRe[f],  false, bIm[f], (short)0, dim_, false, false);
      dim_ = __builtin_amdgcn_wmma_f32_16x16x4_f32(false, aIm[f],  false, bRe[f], (short)0, dim_, false, false);
    }
    #pragma unroll
    for (int v = 0; v < 8; ++v) {
      float2 st; st.x = dre[v]; st.y = dim_[v];
      sAmp[pob ^ pm[v]] = st;                      // ds_store_b64
    }
  }
}

// ---------------------------------------------------------------------------
// Main kernel: NS samples per workgroup, interleaved (re,im) state in LDS.
// ---------------------------------------------------------------------------
__global__ void __launch_bounds__(THREADS)
qsim_kernel(const float* __restrict__ x, const float* __restrict__ head_w,
            const float* __restrict__ head_b, const float* __restrict__ Ure,
            const float* __restrict__ Uim, float* __restrict__ out) {
  extern __shared__ float smem_raw[];
  float2* sAmp = (float2*)smem_raw;                 // NS*DIM float2
  float*  accS = (float*)(sAmp + NS * DIM);         // NS
  float*  csA  = accS + NS;                         // NS*NQ cos(x/2)
  float*  snA  = csA + NS * NQ;                     // NS*NQ sin(x/2)

  const int tid  = threadIdx.x;
  const int lane = tid & 31;
  const int wave = tid >> 5;
  const int s0   = blockIdx.x * NS;

  float w[NQ];
  #pragma unroll
  for (int i = 0; i < NQ; ++i) w[i] = head_w[i];

  // --- Encoding: product state, amp(z) = m(z) * (-i)^popc(z) ---
  if (tid < NS * NQ) {
    const int ss = tid / NQ, i = tid - ss * NQ;
    const float v = 0.5f * x[(s0 + ss) * NQ + i];
    csA[tid] = cosf(v);
    snA[tid] = sinf(v);
  }
  __syncthreads();
  for (int ss = 0; ss < NS; ++ss) {
    for (int z = tid; z < DIM; z += THREADS) {
      float m = 1.f;
      #pragma unroll
      for (int i = 0; i < NQ; ++i)
        m *= ((z >> (NQ - 1 - i)) & 1) ? snA[ss * NQ + i] : csA[ss * NQ + i];
      const int k = __popc(z) & 3;
      float2 a; a.x = 0.f; a.y = 0.f;
      if (k == 0) a.x = m; else if (k == 1) a.y = -m;
      else if (k == 2) a.x = -m; else a.y = m;
      sAmp[(ss << 12) + z] = a;
    }
  }
  __syncthreads();

  // --- 3 layers x 3 groups, fully unrolled so permutations fold ---
  apply_group<0, 0>(sAmp, Ure, Uim, lane, wave); __syncthreads();
  apply_group<0, 1>(sAmp, Ure, Uim, lane, wave); __syncthreads();
  apply_group<0, 2>(sAmp, Ure, Uim, lane, wave); __syncthreads();
  apply_group<1, 0>(sAmp, Ure, Uim, lane, wave); __syncthreads();
  apply_group<1, 1>(sAmp, Ure, Uim, lane, wave); __syncthreads();
  apply_group<1, 2>(sAmp, Ure, Uim, lane, wave); __syncthreads();
  apply_group<2, 0>(sAmp, Ure, Uim, lane, wave); __syncthreads();
  apply_group<2, 1>(sAmp, Ure, Uim, lane, wave); __syncthreads();
  apply_group<2, 2>(sAmp, Ure, Uim, lane, wave); __syncthreads();

  // --- Readout: out[s] = sum_z |amp|^2 * (sum_i w_i (1-2 bit_i)) + b ---
  if (tid < NS) accS[tid] = 0.f;
  __syncthreads();
  float wsum = 0.f;
  #pragma unroll
  for (int i = 0; i < NQ; ++i) wsum += w[i];
  for (int ss = 0; ss < NS; ++ss) {
    float a = 0.f;
    for (int z = tid; z < DIM; z += THREADS) {
      const float2 q = sAmp[(ss << 12) + permc<NL>(z)]; // final pending perm
      float coef = wsum;
      #pragma unroll
      for (int i = 0; i < NQ; ++i)
        coef -= 2.f * w[i] * (float)((z >> (NQ - 1 - i)) & 1);
      a += (q.x * q.x + q.y * q.y) * coef;
    }
    atomicAdd(&accS[ss], a);
  }
  __syncthreads();
  if (tid < NS) out[s0 + tid] = accS[tid] + head_b[0];
}

// ---------------------------------------------------------------------------
extern "C" void kernel_launch(void* const* d_in, const int* in_sizes, int n_in,
                              void* d_out, int out_size, void* d_ws, size_t ws_size,
                              hipStream_t stream) {
  const float* x      = (const float*)d_in[0];   // [B, 12]
  const float* params = (const float*)d_in[1];   // [3, 12, 3]
  const float* head_w = (const float*)d_in[2];   // [1, 12]
  const float* head_b = (const float*)d_in[3];   // [1]
  float* out = (float*)d_out;
  float* Ure = (float*)d_ws;                     // 3*3*256 floats
  float* Uim = Ure + NL * 3 * 256;

  const int B = in_sizes[0] / NQ;                // 8192
  const size_t shmem =
      (size_t)(2 * NS * DIM + NS + 2 * NS * NQ) * sizeof(float);  // ~131 KB
  (void)hipFuncSetAttribute((const void*)qsim_kernel,
                            hipFuncAttributeMaxDynamicSharedMemorySize,
                            (int)shmem);

  qgates_kernel<<<1, 256, 0, stream>>>(params, Ure, Uim);
  qsim_kernel<<<B / NS, THREADS, shmem, stream>>>(x, head_w, head_b, Ure, Uim, out);
}